// CrossScan3D_26319559590241
// MI455X (gfx1250) — compile-verified
//
#include <hip/hip_runtime.h>
#include <stdint.h>

// CrossScan3D for MI455X (gfx1250).
// x: [B=2, C=96, 32,32,32] f32 -> out: [B, 12, C, 32768] f32.
// Pure permutation: HBM-store bound (~302 MB out @ 23.3 TB/s ~ 14 us floor).
// Strategy: per (b,c) slab (128 KB) -> padded LDS via CDNA5 async
// global->LDS b128 loads (ASYNCcnt), then 12 coalesced NT b128 store streams
// gathered from LDS. 4 workgroups share a slab (3 directions each) to fill
// the chip; redundant slab reads come from L2 (input 25 MB << 192 MB L2).

#define L3   32768   // 32*32*32
#define NC   96
#define NB   2

typedef float v4f __attribute__((ext_vector_type(4)));

__host__ __device__ constexpr int padidx(int f) {
  // +1 dword per 32, +1 per 1024: makes lane strides 1/32/1024 conflict-free
  return f + (f >> 5) + (f >> 10);
}

#define SMEM_DWORDS (padidx(L3 - 1) + 1)   // 33822 dwords = 135288 bytes

__global__ void __launch_bounds__(256)
cross_scan3d_kernel(const float* __restrict__ x, float* __restrict__ out) {
  extern __shared__ float smem[];
  const int tid  = threadIdx.x;                 // 0..255
  const int slab = (int)blockIdx.x >> 2;        // 0..191 = b*96 + c
  const int part = (int)blockIdx.x & 3;         // directions part*3 .. part*3+2

  // ---- Phase 1: async-copy the whole 32^3 slab into padded LDS ----------
  {
    const uint64_t gbase = (uint64_t)(uintptr_t)(x + (size_t)slab * L3);
    // Low 32 bits of a flat LDS pointer are the LDS byte offset (aperture
    // base has zero low dword), so truncation yields the VDST address.
    const uint32_t lbase = (uint32_t)(uintptr_t)&smem[0];
#pragma unroll
    for (int it = 0; it < 32; ++it) {
      const int f = it * 1024 + tid * 4;              // dword idx, 16B chunk
      const uint32_t voff  = (uint32_t)(f * 4);       // global byte offset
      const uint32_t laddr = lbase + (uint32_t)(padidx(f) * 4);
      // GVS mode: mem = SGPR64 + VGPR32 ; LDS dest per-lane from VDST VGPR.
      asm volatile("global_load_async_to_lds_b128 %0, %1, %2"
                   :: "v"(laddr), "v"(voff), "s"(gbase)
                   : "memory");
    }
    asm volatile("s_wait_asynccnt 0" ::: "memory");
  }
  __syncthreads();

  // ---- Phase 2: 3 coalesced permuted output streams ----------------------
  const int b     = slab / NC;
  const int c     = slab % NC;
  const int cbase = (tid * 4) & 31;   // c-digit of first of 4 lanes' elements
  const int bdig  = (tid * 4) >> 5;   // b-digit (constant across the 4)

  for (int dd = 0; dd < 3; ++dd) {
    const int  d    = part * 3 + dd;        // 0..11
    const int  base = d & ~2;               // {0,1,4,5,8,9}
    const bool flip = (d & 2) != 0;         // reversed variants
    // out[l] = in[a*ca + b2*cb + c2*cc] with (a,b2,c2) = base-32 digits of l
    int ca, cb, cc;
    switch (base) {
      case 0:  ca = 1024; cb = 32;   cc = 1;    break;  // x0: (i,j,k)
      case 1:  ca = 1024; cb = 1;    cc = 32;   break;  // x1: (i,k,j)
      case 4:  ca = 32;   cb = 1;    cc = 1024; break;  // y0: (j,k,i)
      case 5:  ca = 32;   cb = 1024; cc = 1;    break;  // y1: (j,i,k)
      case 8:  ca = 1;    cb = 1024; cc = 32;   break;  // z0: (k,i,j)
      default: ca = 1;    cb = 32;   cc = 1024; break;  // z1: (k,j,i)
    }
    float* outp = out + ((size_t)((b * 12 + d) * NC + c)) * (size_t)L3;
#pragma unroll 4
    for (int it = 0; it < 32; ++it) {
      const int g0   = it * ca + bdig * cb + cbase * cc;
      const int f0   = flip ? (L3 - 1) - g0 : g0;      // digit complement
      const int step = flip ? -cc : cc;
      v4f v;
      v.x = smem[padidx(f0)];
      v.y = smem[padidx(f0 + step)];
      v.z = smem[padidx(f0 + 2 * step)];
      v.w = smem[padidx(f0 + 3 * step)];
      const int l0 = it * 1024 + tid * 4;
      // NT store: don't let the 302 MB output stream thrash L2 (input lives there)
      __builtin_nontemporal_store(v, (v4f*)(outp + l0));
    }
  }
}

extern "C" void kernel_launch(void* const* d_in, const int* in_sizes, int n_in,
                              void* d_out, int out_size, void* d_ws, size_t ws_size,
                              hipStream_t stream) {
  (void)in_sizes; (void)n_in; (void)d_ws; (void)ws_size; (void)out_size;
  const float* x = (const float*)d_in[0];
  float* out = (float*)d_out;
  const dim3 grid(NB * NC * 4);   // 768 workgroups
  const dim3 block(256);          // 8 wave32s
  const size_t shmem = (size_t)SMEM_DWORDS * sizeof(float);  // 135288 B (<320KB/WGP)
  cross_scan3d_kernel<<<grid, block, shmem, stream>>>(x, out);
}